// NsectCudaLoss_35158602285818
// MI455X (gfx1250) — compile-verified
//
#include <hip/hip_runtime.h>
#include <stdint.h>

// 1.5-entmax n-section loss for MI455X (gfx1250).
//
// Strategy (HBM-roofline bound):
//  - one 1024-thread (32-wave) workgroup per row
//  - row (128 KB) staged HBM -> LDS once via CDNA5 async DMA
//    (global_load_async_to_lds_b128 + s_wait_asynccnt)
//  - each thread keeps its 32 elements (x*0.5) in VGPRs
//  - block max; only elements with xs >= max-1 can ever have relu(xs-tau)>0,
//    so compact those (~1.6% of 32000 for N(0,1)) into LDS via a
//    deterministic prefix-sum scan; all 9x3 f-evals + final pass run over the
//    compacted set only  => kernel is limited by the single 512 MB HBM read.
//  - deterministic reductions everywhere (fixed-order shuffles/LDS trees).

#define BLOCK   1024
#define NWAVES  (BLOCK / 32)
#define KMAX    32           // elements per thread (covers C <= 32768)
#define LDSROW  32000        // row capacity in LDS (C == 32000)

__device__ __forceinline__ void async_ld_b128(unsigned lds_off, const float* g) {
  // CDNA5 async global->LDS DMA, 16 bytes per active lane. LDS dest address is
  // a per-lane byte offset into the workgroup's LDS allocation (VDST operand).
  asm volatile("global_load_async_to_lds_b128 %0, %1, off"
               :: "v"(lds_off), "v"(g)
               : "memory");
}

__device__ __forceinline__ void wait_async0() {
#if __has_builtin(__builtin_amdgcn_s_wait_asynccnt)
  __builtin_amdgcn_s_wait_asynccnt(0);
#else
  asm volatile("s_wait_asynccnt 0" ::: "memory");
#endif
}

// Deterministic 3-value block sum; broadcasts results to all threads.
// Uses 2 shuffle butterflies (wave32) + one LDS stage across 32 waves.
__device__ __forceinline__ void block_reduce3(float& a, float& b, float& c,
                                              float* s_red, float* s_bc,
                                              int lane, int wid) {
#pragma unroll
  for (int d = 16; d >= 1; d >>= 1) {
    a += __shfl_xor(a, d, 32);
    b += __shfl_xor(b, d, 32);
    c += __shfl_xor(c, d, 32);
  }
  if (lane == 0) { s_red[wid * 3 + 0] = a; s_red[wid * 3 + 1] = b; s_red[wid * 3 + 2] = c; }
  __syncthreads();
  if (wid == 0) {
    float x = (lane < NWAVES) ? s_red[lane * 3 + 0] : 0.0f;
    float y = (lane < NWAVES) ? s_red[lane * 3 + 1] : 0.0f;
    float z = (lane < NWAVES) ? s_red[lane * 3 + 2] : 0.0f;
#pragma unroll
    for (int d = 16; d >= 1; d >>= 1) {
      x += __shfl_xor(x, d, 32);
      y += __shfl_xor(y, d, 32);
      z += __shfl_xor(z, d, 32);
    }
    if (lane == 0) { s_bc[0] = x; s_bc[1] = y; s_bc[2] = z; }
  }
  __syncthreads();
  a = s_bc[0]; b = s_bc[1]; c = s_bc[2];
  __syncthreads();  // make s_red / s_bc safe for the next call
}

__global__ __launch_bounds__(BLOCK)
void entmax15_nsect_row_kernel(const float* __restrict__ X,
                               const int*   __restrict__ target,
                               float* __restrict__ row_loss,
                               int C) {
  __shared__ float s_row[LDSROW];       // staged row, then compacted actives
  __shared__ float s_red[NWAVES * 3];
  __shared__ float s_bc[3];
  __shared__ int   s_wsum[NWAVES];
  __shared__ int   s_nact;

  const int tid  = threadIdx.x;
  const int lane = tid & 31;
  const int wid  = tid >> 5;
  const int row  = blockIdx.x;
  const float* Xrow = X + (size_t)row * (size_t)C;

  // ---- Stage row into LDS with async 128-bit DMA (coalesced, read-once) ----
  const unsigned s_row_base = (unsigned)(uintptr_t)(void*)s_row;  // LDS byte offset
  const int C4 = C >> 2;
  for (int q = tid; q < C4; q += BLOCK)
    async_ld_b128(s_row_base + (unsigned)q * 16u, Xrow + q * 4);
  wait_async0();
  for (int i = (C4 << 2) + tid; i < C; i += BLOCK)  // scalar tail (C%4)
    s_row[i] = Xrow[i];
  __syncthreads();

  // ---- Private register copy of xs = 0.5*X (pad with -inf-ish) ----
  float xs[KMAX];
#pragma unroll
  for (int k = 0; k < KMAX; ++k) {
    int idx = tid + k * BLOCK;
    xs[k] = (idx < C) ? s_row[idx] * 0.5f : -1.0e30f;
  }

  // ---- Block max of xs ----
  float m = -1.0e30f;
#pragma unroll
  for (int k = 0; k < KMAX; ++k) m = fmaxf(m, xs[k]);
#pragma unroll
  for (int d = 16; d >= 1; d >>= 1) m = fmaxf(m, __shfl_xor(m, d, 32));
  if (lane == 0) s_red[wid] = m;
  __syncthreads();
  if (wid == 0) {
    float t = (lane < NWAVES) ? s_red[lane] : -1.0e30f;
#pragma unroll
    for (int d = 16; d >= 1; d >>= 1) t = fmaxf(t, __shfl_xor(t, d, 32));
    if (lane == 0) s_bc[0] = t;
  }
  __syncthreads();
  const float mx = s_bc[0];
  __syncthreads();

  // ---- Deterministic compaction of active elements (xs >= mx-1) ----
  // tau in [mx-1, mx] forever => inactive elements contribute exactly 0.
  const float thresh = mx - 1.0f;
  int cnt = 0;
#pragma unroll
  for (int k = 0; k < KMAX; ++k) cnt += (xs[k] >= thresh) ? 1 : 0;

  int incl = cnt;                       // wave32 inclusive scan
#pragma unroll
  for (int d = 1; d < 32; d <<= 1) {
    int t = __shfl_up(incl, d, 32);
    if (lane >= d) incl += t;
  }
  if (lane == 31) s_wsum[wid] = incl;
  __syncthreads();
  if (wid == 0) {
    int w  = (lane < NWAVES) ? s_wsum[lane] : 0;
    int iw = w;
#pragma unroll
    for (int d = 1; d < 32; d <<= 1) {
      int t = __shfl_up(iw, d, 32);
      if (lane >= d) iw += t;
    }
    if (lane < NWAVES) s_wsum[lane] = iw - w;   // exclusive wave base
    if (lane == 31) s_nact = iw;
  }
  __syncthreads();
  int pos = s_wsum[wid] + (incl - cnt);
  const int nact = s_nact;
  __syncthreads();
#pragma unroll
  for (int k = 0; k < KMAX; ++k) {
    if (xs[k] >= thresh) { s_row[pos] = xs[k]; ++pos; }  // row buffer is dead; reuse
  }
  __syncthreads();

  // ---- 9 iterations of 4-section search on the compacted set ----
  float lo = mx - 1.0f;
  float hi = mx;
#pragma unroll 1
  for (int it = 0; it < 9; ++it) {
    const float width = (hi - lo) * 0.25f;
    const float t1 = lo + 1.0f * width;
    const float t2 = lo + 2.0f * width;
    const float t3 = lo + 3.0f * width;
    float a = 0.0f, b = 0.0f, c = 0.0f;
    for (int i = tid; i < nact; i += BLOCK) {
      float x  = s_row[i];
      float r1 = fmaxf(x - t1, 0.0f); a = fmaf(r1, r1, a);
      float r2 = fmaxf(x - t2, 0.0f); b = fmaf(r2, r2, b);
      float r3 = fmaxf(x - t3, 0.0f); c = fmaf(r3, r3, c);
    }
    block_reduce3(a, b, c, s_red, s_bc, lane, wid);
    // f(tau) >= 0  <=>  sum >= 1
    float count = (a >= 1.0f ? 1.0f : 0.0f) + (b >= 1.0f ? 1.0f : 0.0f) +
                  (c >= 1.0f ? 1.0f : 0.0f);
    lo = lo + width * count;
    hi = lo + width;
  }

  // ---- Final pass: S=sum r^2, T=sum r^3, U=sum r^2*xs ----
  const float tau = 0.5f * (lo + hi);
  float S = 0.0f, T = 0.0f, U = 0.0f;
  for (int i = tid; i < nact; i += BLOCK) {
    float x = s_row[i];
    float r = fmaxf(x - tau, 0.0f);
    float p = r * r;
    S += p;
    T = fmaf(p, r, T);
    U = fmaf(p, x, U);
  }
  block_reduce3(S, T, U, s_red, s_bc, lane, wid);

  if (tid == 0) {
    int   tgt = target[row];
    float xt  = Xrow[tgt];
    // omega = (1 - sum (p/S)^1.5)/0.75 = (1 - T/S^1.5)/0.75
    float omega = (1.0f - T / (S * sqrtf(S))) / 0.75f;
    // sum(p_norm * X) = 2*U/S  (X = 2*xs);  p_star subtracts e_target
    float dotpx = 2.0f * U / S;
    row_loss[row] = omega + dotpx - xt;
  }
}

// Deterministic fixed-order final reduction of per-row losses -> mean.
__global__ __launch_bounds__(BLOCK)
void finalize_kernel(const float* __restrict__ row_loss, float* __restrict__ out, int n) {
  __shared__ float s[BLOCK];
  int tid = threadIdx.x;
  float v = 0.0f;
  for (int i = tid; i < n; i += BLOCK) v += row_loss[i];
  s[tid] = v;
  __syncthreads();
  for (int step = BLOCK / 2; step > 0; step >>= 1) {
    if (tid < step) s[tid] += s[tid + step];
    __syncthreads();
  }
  if (tid == 0) out[0] = s[0] / (float)n;
}

extern "C" void kernel_launch(void* const* d_in, const int* in_sizes, int n_in,
                              void* d_out, int out_size, void* d_ws, size_t ws_size,
                              hipStream_t stream) {
  const float* X      = (const float*)d_in[0];
  const int*   target = (const int*)d_in[1];
  float*       out    = (float*)d_out;
  float*       ws     = (float*)d_ws;   // n floats of per-row loss

  const int n = in_sizes[1];
  const int C = in_sizes[0] / n;

  entmax15_nsect_row_kernel<<<n, BLOCK, 0, stream>>>(X, target, ws, C);
  finalize_kernel<<<1, BLOCK, 0, stream>>>(ws, out, n);
}